// pmformer_49108656062716
// MI455X (gfx1250) — compile-verified
//
#include <hip/hip_runtime.h>
#include <hip/hip_bf16.h>

typedef float v8f __attribute__((ext_vector_type(8)));
typedef float v2f __attribute__((ext_vector_type(2)));

// ---------------- problem constants ----------------
constexpr int NPTS   = 250000;
constexpr int NTILES = NPTS / 16;          // 15625 (exact)
// per-scale grids: ps = {2,4,6,8}
constexpr int PSA[4] = {2, 4, 6, 8};
constexpr int GXA[4] = {128, 64, 43, 32};
constexpr int GYA[4] = {128, 64, 43, 32};
constexpr int GZA[4] = {16, 8, 6, 4};
// segment offsets (segment units) and total
constexpr int SOFF[5] = {0, 524288, 589824, 612012, 620204};
constexpr int NSEG_TOT = 620204;

// workspace layout (float units)
constexpr size_t RED_OFF  = 0;
constexpr size_t SUM_OFF  = (size_t)NPTS * 64;                 // 16,000,000
constexpr size_t CNT_OFF  = SUM_OFF + (size_t)NSEG_TOT * 64;   // +39,693,056
constexpr size_t PROJ_OFF = CNT_OFF + 620224;                  // cnt padded

__device__ __forceinline__ v8f v8f_splat(float v) {
  v8f r;
#pragma unroll
  for (int i = 0; i < 8; ++i) r[i] = v;
  return r;
}

__device__ __forceinline__ v8f wmma4(v2f a, v2f b, v8f c) {
  // D = A(16x4 f32) * B(4x16 f32) + C(16x16 f32)
  return __builtin_amdgcn_wmma_f32_16x16x4_f32(
      false, a, false, b, (short)0, c, false, false);
}

__device__ __forceinline__ float sigm(float x) {
  return 1.0f / (1.0f + __expf(-x));
}

__device__ __forceinline__ int seg_of(int b, int x, int y, int z, int j) {
  int qx = x / PSA[j], qy = y / PSA[j], qz = z / PSA[j];
  return ((b * GXA[j] + qx) * GYA[j] + qy) * GZA[j] + qz;
}

// ---------------- kernel A: red = relu(x @ W_red + b_red) ----------------
__global__ void kA(const float* __restrict__ x, const float* __restrict__ Wr,
                   const float* __restrict__ br, float* __restrict__ red) {
  int wave = threadIdx.x >> 5, l = threadIdx.x & 31;
  int tile = blockIdx.x * 4 + wave;
  if (tile >= NTILES) return;
  int m16 = l & 15, kh = l >> 4;
  const float* xr = x + (size_t)(tile * 16 + m16) * 64;
  v8f acc[4];
#pragma unroll
  for (int ct = 0; ct < 4; ++ct) acc[ct] = v8f_splat(br[ct * 16 + m16]);
#pragma unroll
  for (int kk = 0; kk < 16; ++kk) {
    int k = 4 * kk + 2 * kh;
    v2f a = *(const v2f*)(xr + k);
#pragma unroll
    for (int ct = 0; ct < 4; ++ct) {
      v2f b;
      b.x = Wr[k * 64 + ct * 16 + m16];
      b.y = Wr[(k + 1) * 64 + ct * 16 + m16];
      acc[ct] = wmma4(a, b, acc[ct]);
    }
  }
#pragma unroll
  for (int ct = 0; ct < 4; ++ct)
#pragma unroll
    for (int r = 0; r < 8; ++r) {
      int row = tile * 16 + r + 8 * kh;
      red[(size_t)row * 64 + ct * 16 + m16] = fmaxf(acc[ct][r], 0.0f);
    }
}

// ---------------- kernel B: scatter red into segment sums/counts ----------------
__global__ void kB(const float* __restrict__ red, const int* __restrict__ coords,
                   float* __restrict__ sums, float* __restrict__ cnt) {
  int t = blockIdx.x * blockDim.x + threadIdx.x;
  if (t >= NPTS * 64) return;
  int p = t >> 6, c = t & 63;
  float v = red[t];
  int cb = coords[p * 4 + 0], cx = coords[p * 4 + 1];
  int cy = coords[p * 4 + 2], cz = coords[p * 4 + 3];
#pragma unroll
  for (int j = 0; j < 4; ++j) {
    int s = SOFF[j] + seg_of(cb, cx, cy, cz, j);
    atomicAdd(&sums[(size_t)s * 64 + c], v);
    if (c == 0) atomicAdd(&cnt[s], 1.0f);
  }
}

// ---------------- kernel C: att = relu(mean @ W_pool[j] + b_pool[j]) in place ----------------
__global__ void kC(float* __restrict__ seg_sums, const float* __restrict__ segcnt,
                   const float* __restrict__ Wp, const float* __restrict__ bp, int ns) {
  int wave = threadIdx.x >> 5, l = threadIdx.x & 31;
  int tile = blockIdx.x * 4 + wave;
  int ntiles = (ns + 15) >> 4;
  if (tile >= ntiles) return;
  int m16 = l & 15, kh = l >> 4;
  int s = tile * 16 + m16;
  if (s >= ns) s = ns - 1;  // clamp; clamped rows masked at store
  float inv = 1.0f / fmaxf(segcnt[s], 1.0f);
  const float* srow = seg_sums + (size_t)s * 64;
  v8f a0 = v8f_splat(bp[m16]);
  v8f a1 = v8f_splat(bp[16 + m16]);
#pragma unroll
  for (int kk = 0; kk < 16; ++kk) {
    int k = 4 * kk + 2 * kh;
    v2f a;
    a.x = srow[k] * inv;
    a.y = srow[k + 1] * inv;
    v2f b0, b1;
    b0.x = Wp[k * 32 + m16];        b0.y = Wp[(k + 1) * 32 + m16];
    b1.x = Wp[k * 32 + 16 + m16];   b1.y = Wp[(k + 1) * 32 + 16 + m16];
    a0 = wmma4(a, b0, a0);
    a1 = wmma4(a, b1, a1);
  }
#pragma unroll
  for (int r = 0; r < 8; ++r) {
    int sg = tile * 16 + r + 8 * kh;
    if (sg < ns) {
      float* d = seg_sums + (size_t)sg * 64;
      d[m16]      = fmaxf(a0[r], 0.0f);
      d[16 + m16] = fmaxf(a1[r], 0.0f);
    }
  }
}

// ---------------- kernel D: fused per-point tail ----------------
__global__ void kD(const float* __restrict__ red, const int* __restrict__ coords,
                   const float* __restrict__ sums, const float* __restrict__ W_z,
                   const float* __restrict__ W_att, const float* __restrict__ b_att,
                   const float* __restrict__ W_out, const float* __restrict__ W_l1,
                   const float* __restrict__ W_l2, const float* __restrict__ b_l2,
                   float* __restrict__ proj) {
  __shared__ int   SEG[2][16][4];
  __shared__ float SF[2][16][128];   // sf: [point][scale*32 + c]
  __shared__ float U[2][16][200];    // staging: padded stride vs 64 banks

  int wave = threadIdx.x >> 5, l = threadIdx.x & 31;
  int tile = blockIdx.x * 2 + wave;
  if (tile >= NTILES) return;
  int m16 = l & 15, kh = l >> 4;
  int base = tile * 16;
  int   (&seg)[16][4]  = SEG[wave];
  float (&sf)[16][128] = SF[wave];
  float (&u)[16][200]  = U[wave];

  // 1) segment ids per point
  if (l < 16) {
    int p = base + l;
    int cb = coords[p * 4 + 0], cx = coords[p * 4 + 1];
    int cy = coords[p * 4 + 2], cz = coords[p * 4 + 3];
#pragma unroll
    for (int j = 0; j < 4; ++j) seg[l][j] = SOFF[j] + seg_of(cb, cx, cy, cz, j);
  }
  __builtin_amdgcn_wave_barrier();

  // 2) gather sf (att vectors live in first 32 slots of each 64-wide segment row)
#pragma unroll
  for (int m = 0; m < 16; ++m)
#pragma unroll
    for (int j = 0; j < 4; ++j)
      sf[m][j * 32 + l] = sums[(size_t)seg[m][j] * 64 + l];
  __builtin_amdgcn_wave_barrier();

  // 3) s = sum_j sf  -> u[:,0:32]
#pragma unroll
  for (int m = 0; m < 16; ++m)
    u[m][l] = sf[m][l] + sf[m][32 + l] + sf[m][64 + l] + sf[m][96 + l];
  __builtin_amdgcn_wave_barrier();

  // 4) feat_Z = relu(s @ W_z)  (32->32) -> u[:,32:64]
  {
    v8f c0 = v8f_splat(0.0f), c1 = v8f_splat(0.0f);
#pragma unroll
    for (int kk = 0; kk < 8; ++kk) {
      int k = 4 * kk + 2 * kh;
      v2f a; a.x = u[m16][k]; a.y = u[m16][k + 1];
      v2f b0, b1;
      b0.x = W_z[k * 32 + m16];       b0.y = W_z[(k + 1) * 32 + m16];
      b1.x = W_z[k * 32 + 16 + m16];  b1.y = W_z[(k + 1) * 32 + 16 + m16];
      c0 = wmma4(a, b0, c0);
      c1 = wmma4(a, b1, c1);
    }
#pragma unroll
    for (int r = 0; r < 8; ++r) {
      int row = r + 8 * kh;
      u[row][32 + m16] = fmaxf(c0[r], 0.0f);
      u[row][48 + m16] = fmaxf(c1[r], 0.0f);
    }
  }
  __builtin_amdgcn_wave_barrier();

  // 5) g = sum_j sf_j * sigmoid(feat_Z @ W_att[j] + b_att[j])
  v8f g0 = v8f_splat(0.0f), g1 = v8f_splat(0.0f);
#pragma unroll
  for (int j = 0; j < 4; ++j) {
    v8f c0 = v8f_splat(b_att[j * 32 + m16]);
    v8f c1 = v8f_splat(b_att[j * 32 + 16 + m16]);
#pragma unroll
    for (int kk = 0; kk < 8; ++kk) {
      int k = 4 * kk + 2 * kh;
      v2f a; a.x = u[m16][32 + k]; a.y = u[m16][32 + k + 1];
      v2f b0, b1;
      b0.x = W_att[(j * 32 + k) * 32 + m16];      b0.y = W_att[(j * 32 + k + 1) * 32 + m16];
      b1.x = W_att[(j * 32 + k) * 32 + 16 + m16]; b1.y = W_att[(j * 32 + k + 1) * 32 + 16 + m16];
      c0 = wmma4(a, b0, c0);
      c1 = wmma4(a, b1, c1);
    }
#pragma unroll
    for (int r = 0; r < 8; ++r) {
      int row = r + 8 * kh;
      g0[r] += sigm(c0[r]) * sf[row][j * 32 + m16];
      g1[r] += sigm(c1[r]) * sf[row][j * 32 + 16 + m16];
    }
  }
  // write g -> u[:,0:32] (s is dead)
#pragma unroll
  for (int r = 0; r < 8; ++r) {
    int row = r + 8 * kh;
    u[row][m16] = g0[r];
    u[row][16 + m16] = g1[r];
  }
  __builtin_amdgcn_wave_barrier();

  // 6) scale_features = g @ W_out (32->64) -> u[:,64:128]
  {
    v8f o[4];
#pragma unroll
    for (int ct = 0; ct < 4; ++ct) o[ct] = v8f_splat(0.0f);
#pragma unroll
    for (int kk = 0; kk < 8; ++kk) {
      int k = 4 * kk + 2 * kh;
      v2f a; a.x = u[m16][k]; a.y = u[m16][k + 1];
#pragma unroll
      for (int ct = 0; ct < 4; ++ct) {
        v2f b;
        b.x = W_out[k * 64 + ct * 16 + m16];
        b.y = W_out[(k + 1) * 64 + ct * 16 + m16];
        o[ct] = wmma4(a, b, o[ct]);
      }
    }
#pragma unroll
    for (int ct = 0; ct < 4; ++ct)
#pragma unroll
      for (int r = 0; r < 8; ++r)
        u[r + 8 * kh][64 + ct * 16 + m16] = o[ct][r];
  }
  __builtin_amdgcn_wave_barrier();

  // 7) load red -> u[:,0:64]  (out_f = [red | scale_features] in u[:,0:128])
#pragma unroll
  for (int it = 0; it < 32; ++it) {
    int flat = it * 32 + l;
    int m = flat >> 6, c = flat & 63;
    u[m][c] = red[(size_t)(base + m) * 64 + c];
  }
  __builtin_amdgcn_wave_barrier();

  // 8) h = relu(out_f @ W_l1) (128->64) -> u[:,128:192]
  {
    v8f h[4];
#pragma unroll
    for (int ct = 0; ct < 4; ++ct) h[ct] = v8f_splat(0.0f);
#pragma unroll
    for (int kk = 0; kk < 32; ++kk) {
      int k = 4 * kk + 2 * kh;
      v2f a; a.x = u[m16][k]; a.y = u[m16][k + 1];
#pragma unroll
      for (int ct = 0; ct < 4; ++ct) {
        v2f b;
        b.x = W_l1[k * 64 + ct * 16 + m16];
        b.y = W_l1[(k + 1) * 64 + ct * 16 + m16];
        h[ct] = wmma4(a, b, h[ct]);
      }
    }
#pragma unroll
    for (int ct = 0; ct < 4; ++ct)
#pragma unroll
      for (int r = 0; r < 8; ++r)
        u[r + 8 * kh][128 + ct * 16 + m16] = fmaxf(h[ct][r], 0.0f);
  }
  __builtin_amdgcn_wave_barrier();

  // 9) proj = h @ W_l2 + b_l2 (64->64) -> global
  {
    v8f o[4];
#pragma unroll
    for (int ct = 0; ct < 4; ++ct) o[ct] = v8f_splat(b_l2[ct * 16 + m16]);
#pragma unroll
    for (int kk = 0; kk < 16; ++kk) {
      int k = 4 * kk + 2 * kh;
      v2f a; a.x = u[m16][128 + k]; a.y = u[m16][128 + k + 1];
#pragma unroll
      for (int ct = 0; ct < 4; ++ct) {
        v2f b;
        b.x = W_l2[k * 64 + ct * 16 + m16];
        b.y = W_l2[(k + 1) * 64 + ct * 16 + m16];
        o[ct] = wmma4(a, b, o[ct]);
      }
    }
#pragma unroll
    for (int ct = 0; ct < 4; ++ct)
#pragma unroll
      for (int r = 0; r < 8; ++r) {
        int row = base + r + 8 * kh;
        proj[(size_t)row * 64 + ct * 16 + m16] = o[ct][r];
      }
  }
}

// ---------------- kernel E: out[i] = proj[coords_inv[i]] ----------------
__global__ void kE(const float* __restrict__ proj, const int* __restrict__ cinv,
                   float* __restrict__ out) {
  int t = blockIdx.x * blockDim.x + threadIdx.x;
  if (t >= NPTS * 64) return;
  int i = t >> 6, c = t & 63;
  out[t] = proj[(size_t)cinv[i] * 64 + c];
}

extern "C" void kernel_launch(void* const* d_in, const int* in_sizes, int n_in,
                              void* d_out, int out_size, void* d_ws, size_t ws_size,
                              hipStream_t stream) {
  (void)in_sizes; (void)n_in; (void)out_size; (void)ws_size;
  const float* x      = (const float*)d_in[0];
  const int*   coords = (const int*)d_in[1];
  const int*   cinv   = (const int*)d_in[2];
  const float* W_red  = (const float*)d_in[3];
  const float* b_red  = (const float*)d_in[4];
  const float* W_pool = (const float*)d_in[5];
  const float* b_pool = (const float*)d_in[6];
  const float* W_att  = (const float*)d_in[7];
  const float* b_att  = (const float*)d_in[8];
  const float* W_z    = (const float*)d_in[9];
  const float* W_out  = (const float*)d_in[10];
  const float* W_l1   = (const float*)d_in[11];
  const float* W_l2   = (const float*)d_in[12];
  const float* b_l2   = (const float*)d_in[13];
  float* out = (float*)d_out;
  float* ws  = (float*)d_ws;

  float* red  = ws + RED_OFF;
  float* sums = ws + SUM_OFF;
  float* cnt  = ws + CNT_OFF;
  float* proj = ws + PROJ_OFF;

  // zero segment sums + counts (graph-capturable)
  hipMemsetAsync(sums, 0, (PROJ_OFF - SUM_OFF) * sizeof(float), stream);

  // red = relu(x @ W_red + b_red): 4 waves/block, wave per 16-row tile
  kA<<<(NTILES + 3) / 4, 128, 0, stream>>>(x, W_red, b_red, red);

  // scatter into segment sums/counts
  kB<<<(NPTS * 64 + 255) / 256, 256, 0, stream>>>(red, coords, sums, cnt);

  // per-scale segment att (in place over sums)
  for (int j = 0; j < 4; ++j) {
    int ns = SOFF[j + 1] - SOFF[j];
    int ntiles = (ns + 15) / 16;
    kC<<<(ntiles + 3) / 4, 128, 0, stream>>>(sums + (size_t)SOFF[j] * 64,
                                             cnt + SOFF[j],
                                             W_pool + (size_t)j * 64 * 32,
                                             b_pool + j * 32, ns);
  }

  // fused per-point tail: 2 waves/block
  kD<<<(NTILES + 1) / 2, 64, 0, stream>>>(red, coords, sums, W_z, W_att, b_att,
                                          W_out, W_l1, W_l2, b_l2, proj);

  // final gather by coords_inv
  kE<<<(NPTS * 64 + 255) / 256, 256, 0, stream>>>(proj, cinv, out);
}